// RNN_2637109920333
// MI455X (gfx1250) — compile-verified
//
#include <hip/hip_runtime.h>
#include <hip/hip_bf16.h>

#define T_STEPS 10
#define HW_N    20736   // 144*144
#define BLK     256

#if __has_builtin(__builtin_amdgcn_tanhf)
__device__ __forceinline__ float fast_tanh(float x) { return __builtin_amdgcn_tanhf(x); }
#else
__device__ __forceinline__ float fast_tanh(float x) { return tanhf(x); }
#endif

// sigmoid(z) = 0.5 * tanh(z/2) + 0.5  -> single v_tanh_f32 per gate on gfx1250
__device__ __forceinline__ float fast_sigmoid(float z) {
    return fmaf(0.5f, fast_tanh(0.5f * z), 0.5f);
}

__global__ __launch_bounds__(BLK) void lstm_cn_kernel(
    const float* __restrict__ x,
    const float* __restrict__ w_ih,
    const float* __restrict__ w_hh,
    const float* __restrict__ b_ih,
    const float* __restrict__ b_hh,
    float* __restrict__ out)
{
    __shared__ float tile[T_STEPS * BLK];   // 10 KB: 10 timesteps x 256 pixels

    const int tid = threadIdx.x;
    const int gid = blockIdx.x * BLK + tid; // identity map: gid = h*W + w

    // ---- Stage all T timesteps for this block's pixels into LDS with the
    //      gfx1250 async-copy path (ASYNCcnt-tracked, CDNA5 ISA).
    //      Each lane loads its own column; no cross-lane sharing, so a single
    //      s_wait_asynccnt 0 (no barrier) orders the LDS reads.
    {
        // low 32 bits of a flat LDS pointer == wave-relative LDS byte offset
        const uint32_t lds_base = (uint32_t)(uintptr_t)(&tile[tid]);
        const float* gp = x + gid;
#pragma unroll
        for (int t = 0; t < T_STEPS; ++t) {
            const uint64_t ga = (uint64_t)(uintptr_t)(gp + t * HW_N);
            const uint32_t la = lds_base + (uint32_t)(t * BLK * 4);
            asm volatile("global_load_async_to_lds_b32 %0, %1, off"
                         :: "v"(la), "v"(ga)
                         : "memory");
        }
    }

    // ---- Uniform scalars (SMEM loads): gate weights and fused bias.
    const float wi0 = w_ih[0], wi1 = w_ih[1], wi2 = w_ih[2], wi3 = w_ih[3];
    const float wh0 = w_hh[0], wh1 = w_hh[1], wh2 = w_hh[2], wh3 = w_hh[3];
    const float b0 = b_ih[0] + b_hh[0];
    const float b1 = b_ih[1] + b_hh[1];
    const float b2 = b_ih[2] + b_hh[2];
    const float b3 = b_ih[3] + b_hh[3];

    // ---- Wait for all async transfers, then run the serial recurrence.
    asm volatile("s_wait_asynccnt 0" ::: "memory");

    float h = 0.0f, c = 0.0f;
#pragma unroll
    for (int t = 0; t < T_STEPS; ++t) {
        const float xt = tile[t * BLK + tid];

        const float zi = fmaf(xt, wi0, fmaf(h, wh0, b0));
        const float zf = fmaf(xt, wi1, fmaf(h, wh1, b1));
        const float zg = fmaf(xt, wi2, fmaf(h, wh2, b2));
        const float zo = fmaf(xt, wi3, fmaf(h, wh3, b3));

        const float ig = fast_sigmoid(zi);
        const float fg = fast_sigmoid(zf);
        const float gg = fast_tanh(zg);
        const float og = fast_sigmoid(zo);

        c = fmaf(fg, c, ig * gg);
        h = og * fast_tanh(c);
    }

    out[gid] = c;
}

extern "C" void kernel_launch(void* const* d_in, const int* in_sizes, int n_in,
                              void* d_out, int out_size, void* d_ws, size_t ws_size,
                              hipStream_t stream) {
    (void)in_sizes; (void)n_in; (void)d_ws; (void)ws_size; (void)out_size;

    const float* x    = (const float*)d_in[0];  // (10,144,144) f32
    const float* w_ih = (const float*)d_in[1];  // (4,1) f32
    const float* w_hh = (const float*)d_in[2];  // (4,1) f32
    const float* b_ih = (const float*)d_in[3];  // (4,)  f32
    const float* b_hh = (const float*)d_in[4];  // (4,)  f32
    float* out = (float*)d_out;                 // 20736 f32

    lstm_cn_kernel<<<HW_N / BLK, BLK, 0, stream>>>(x, w_ih, w_hh, b_ih, b_hh, out);
}